// TDTLossNumba_40638980555328
// MI455X (gfx1250) — compile-verified
//
#include <hip/hip_runtime.h>

// ---------------- problem constants (mirror the reference) ----------------
constexpr int   cB = 4, cT = 256, cU = 101, cV = 1024, cD = 5;
constexpr int   cBLANK = 1023;
constexpr float cSIGMA = 0.05f;
constexpr float NEG_INF = -1.0e30f;

// Workspace bundle layout: one 1024-float (4096 B) bundle per (b,t).
//   [E0_OFF + u]              : e0[u]   = y_lp + dlp0          (u in 0..100)
//   [SB_OFF + 4u + (i-1)]     : sb[u,i] = blank_lp + dlp_i     (i in 1..4)
//   [SY_OFF + 4u + (i-1)]     : sy[u,i] = y_lp + dlp_i
constexpr int RS = 1024;        // bundle stride in floats (4096 B)
constexpr int E0_OFF = 0;
constexpr int SB_OFF = 104;     // 104 + 404 = 508  <= 512
constexpr int SY_OFF = 512;     // 512 + 404 = 916  <= 1024

#if defined(__gfx1250__) && __has_builtin(__builtin_amdgcn_tensor_load_to_lds) && \
    __has_builtin(__builtin_amdgcn_s_wait_tensorcnt)
#define HAVE_TDM 1
#endif

typedef unsigned int u32x4 __attribute__((ext_vector_type(4)));
typedef int          i32x4 __attribute__((ext_vector_type(4)));
typedef int          i32x8 __attribute__((ext_vector_type(8)));

// ---------------- numerically stable logaddexp ----------------
__device__ __forceinline__ float lax(float a, float b) {
  float m = fmaxf(a, b);
  float d = fabsf(a - b);
  // both ~ -1e30 -> d = 0 -> m + log(2) == m in fp32. never NaN for finite in.
  return m + log1pf(expf(-d));
}

// =====================================================================
// Kernel A: streaming log-softmax over acts (memory-bound, 426 MB read).
// One wave32 per (b,t,u) row of 1029 floats.
// =====================================================================
__global__ void __launch_bounds__(256)
tdt_softmax_pack(const float* __restrict__ acts,
                 const int*   __restrict__ labels,
                 float*       __restrict__ wsb) {
  const int lane = threadIdx.x & 31;
  const int wid  = (int)((blockIdx.x * blockDim.x + threadIdx.x) >> 5);
  const int nw   = (int)((gridDim.x * blockDim.x) >> 5);
  const int R    = cB * cT * cU;

  for (int r = wid; r < R; r += nw) {
    const float* a = acts + (size_t)r * (cV + cD);

    // prefetch the next row this wave will touch (gfx1250: global_prefetch_b8)
    if (r + nw < R) {
      const char* nx = (const char*)(acts + (size_t)(r + nw) * (cV + cD));
      __builtin_prefetch((const void*)(nx + lane * 132), 0, 1);
    }

    const int b   = r / (cT * cU);
    const int rem = r - b * (cT * cU);
    const int t   = rem / cU;
    const int u   = rem - t * cU;

    // ---- load 1024 label logits: 32 per lane (8 chunks x float4-contig) ----
    float v[32];
#pragma unroll
    for (int c = 0; c < 8; ++c) {
      const int e = (c * 32 + lane) * 4;
      v[4 * c + 0] = a[e + 0];
      v[4 * c + 1] = a[e + 1];
      v[4 * c + 2] = a[e + 2];
      v[4 * c + 3] = a[e + 3];
    }
    float m = v[0];
#pragma unroll
    for (int k = 1; k < 32; ++k) m = fmaxf(m, v[k]);
#pragma unroll
    for (int off = 16; off > 0; off >>= 1)
      m = fmaxf(m, __shfl_xor(m, off, 32));

    float s = 0.f;
#pragma unroll
    for (int k = 0; k < 32; ++k) s += __expf(v[k] - m);
#pragma unroll
    for (int off = 16; off > 0; off >>= 1)
      s += __shfl_xor(s, off, 32);

    const float lse = m + logf(s);

    // ---- duration log-softmax over 5 logits (redundant per lane, cheap) ----
    const float d0 = a[cV + 0], d1 = a[cV + 1], d2 = a[cV + 2],
                d3 = a[cV + 3], d4 = a[cV + 4];
    float dm = fmaxf(fmaxf(fmaxf(d0, d1), fmaxf(d2, d3)), d4);
    float dls = dm + logf(__expf(d0 - dm) + __expf(d1 - dm) + __expf(d2 - dm) +
                          __expf(d3 - dm) + __expf(d4 - dm));

    const int lab  = (u < cU - 1) ? labels[b * (cU - 1) + u] : 0;
    const float ylp = a[lab]    - lse - cSIGMA;
    const float blp = a[cBLANK] - lse - cSIGMA;

    float* bundle = wsb + (size_t)(b * cT + t) * RS;
    if (lane == 8) {
      bundle[E0_OFF + u] = ylp + (d0 - dls);
    } else if (lane < 8) {
      const int j  = (lane < 4) ? (lane + 1) : (lane - 3);   // duration index 1..4
      const float dj = (j == 1) ? d1 : (j == 2) ? d2 : (j == 3) ? d3 : d4;
      const float dlpj = dj - dls;
      if (lane < 4) bundle[SB_OFF + 4 * u + lane]       = blp + dlpj;
      else          bundle[SY_OFF + 4 * u + (lane - 4)] = ylp + dlpj;
    }
  }
}

// =====================================================================
// Async bundle staging (ASYNCcnt path): 4096 B global -> LDS via
// 8x global_load_async_to_lds_b128 (32 lanes x 16 B each). In-order done.
// =====================================================================
__device__ __forceinline__ void issue_bundle_async(const float* gbase,
                                                   float* lds_ptr, int lane) {
#if defined(__gfx1250__)
  // Flat LDS addresses carry the LDS byte offset in the low 32 bits; the async
  // unit adds the wave's LDS_BASE itself.
  unsigned           l = (unsigned)(unsigned long long)lds_ptr + (unsigned)lane * 16u;
  unsigned long long g = (unsigned long long)gbase + (unsigned long long)lane * 16ull;
#pragma unroll
  for (int c = 0; c < 8; ++c) {
    asm volatile("global_load_async_to_lds_b128 %0, %1, off"
                 :: "v"(l), "v"(g)
                 : "memory");
    g += 512ull;   // 32 lanes * 16 B
    l += 512u;
  }
#else
  for (int c = 0; c < 8; ++c) {
    const int e = (c * 32 + lane) * 4;
    lds_ptr[e + 0] = gbase[e + 0];
    lds_ptr[e + 1] = gbase[e + 1];
    lds_ptr[e + 2] = gbase[e + 2];
    lds_ptr[e + 3] = gbase[e + 3];
  }
#endif
}

__device__ __forceinline__ void wait_async_bundles(int bundles_after) {
#if defined(__gfx1250__)
  // 8 async instructions per bundle; ASYNCcnt completes in-order, so waiting
  // to <= 8*bundles_after guarantees the oldest bundle has fully landed.
  if (bundles_after >= 3)      asm volatile("s_wait_asynccnt 0x18" ::: "memory");
  else if (bundles_after == 2) asm volatile("s_wait_asynccnt 0x10" ::: "memory");
  else if (bundles_after == 1) asm volatile("s_wait_asynccnt 0x8"  ::: "memory");
  else                         asm volatile("s_wait_asynccnt 0x0"  ::: "memory");
#else
  (void)bundles_after;
#endif
}

// =====================================================================
// TDM bundle staging (TENSORcnt path): one Tensor-DMA descriptor moves the
// whole 4 KB bundle as a 1-D tile (1024 x 4 B), one done-signal per bundle.
// D# packed per CDNA5 ISA ch.8: group0 {count=1, lds_addr, global_addr,
// type=2}; group1 {data_size=4B, tensor_dim0=tile_dim0=1024, stride=1024};
// groups 2/3 dims = 1 so no OOB drop. workgroup_mask=0 (not in a cluster).
// =====================================================================
__device__ __forceinline__ void issue_bundle_tdm(const float* gbase,
                                                 float* lds_ptr, int lane) {
#if defined(HAVE_TDM)
  (void)lane;
  unsigned long long ga = (unsigned long long)gbase;        // byte address
  unsigned lds = (unsigned)(unsigned long long)lds_ptr;     // LDS byte offset

  u32x4 g0;
  g0[0] = 0x1u;                                  // count=1, user descriptor
  g0[1] = lds;                                   // lds_addr (bytes)
  g0[2] = (unsigned)(ga & 0xFFFFFFFFu);          // global_addr[31:0]
  g0[3] = (unsigned)((ga >> 32) & 0x01FFFFFFu)   // global_addr[56:32]
        | (2u << 30);                            // type = 2 ("image")

  i32x8 g1;
  g1[0] = 0x00020000;          // workgroup_mask=0, data_size=2 (4 bytes)
  g1[1] = (int)(1024u << 16);  // tensor_dim0 = 1024 (bits 79:48, low half)
  g1[2] = 0x00010000;          // tensor_dim0 hi=0; tensor_dim1 = 1
  g1[3] = (int)(1024u << 16);  // tensor_dim1 hi=0; tile_dim0 = 1024
  g1[4] = 0;                   // tile_dim1 = 0 (unused), tile_dim2 = 0
  g1[5] = 1024;                // tensor_dim0_stride = 1024 elements
  g1[6] = 0;                   // stride hi, tensor_dim1_stride lo
  g1[7] = 0;

  i32x4 g2; g2[0] = 1; g2[1] = 1; g2[2] = 0; g2[3] = 0;  // tensor_dim2/3 = 1
  i32x4 g3; g3[0] = 0; g3[1] = 0x00010000; g3[2] = 0; g3[3] = 0; // tensor_dim4=1

#if __clang_major__ >= 23
  i32x8 g4 = {0, 0, 0, 0, 0, 0, 0, 0};
  __builtin_amdgcn_tensor_load_to_lds(g0, g1, g2, g3, g4, 0);
#else
  __builtin_amdgcn_tensor_load_to_lds(g0, g1, g2, g3, 0);
#endif
#else
  issue_bundle_async(gbase, lds_ptr, lane);
#endif
}

__device__ __forceinline__ void wait_tensor_bundles(int bundles_after) {
#if defined(HAVE_TDM)
  // one TENSORcnt per bundle; in-order completion among tensor ops.
  if (bundles_after >= 3)      __builtin_amdgcn_s_wait_tensorcnt(3);
  else if (bundles_after == 2) __builtin_amdgcn_s_wait_tensorcnt(2);
  else if (bundles_after == 1) __builtin_amdgcn_s_wait_tensorcnt(1);
  else                         __builtin_amdgcn_s_wait_tensorcnt(0);
#else
  wait_async_bundles(bundles_after);
#endif
}

// =====================================================================
// Kernel B: the TDT forward DP. One block (128 threads) per batch element.
// Rolling LDS alpha window (5 rows); bundles prefetched 3 steps ahead:
// prologue bundles 0..2 via ASYNCcnt path, steady state via TDM.
// =====================================================================
__global__ void __launch_bounds__(128)
tdt_dp(const float* __restrict__ wsb,
       const int*   __restrict__ act_lens,
       const int*   __restrict__ label_lens,
       float*       __restrict__ partial) {
  const int b    = blockIdx.x;
  const int tid  = threadIdx.x;
  const int u    = tid;
  const int lane = tid & 31;

  __shared__ float stage[8][RS];       // 8 x 4 KB rotating bundle slots
  __shared__ float alpha_s[5][128];    // rolling alpha rows (t-4..t)
  __shared__ float sc[2][128];         // scan ping-pong
  __shared__ float term[4];            // final duration terms

  if (tid < 4) term[tid] = NEG_INF;

  const float* gB  = wsb + (size_t)b * cT * RS;
  const int alen   = act_lens[b];
  const int Lb     = label_lens[b];

  // prologue: bundles 0,1,2 via the async-to-LDS path (wave 0 only)
  if (tid < 32) {
    for (int pb = 0; pb < 3; ++pb)
      issue_bundle_async(gB + (size_t)pb * RS, &stage[pb & 7][0], lane);
  }
  __syncthreads();

  for (int t = 0; t < cT; ++t) {
    if (tid < 32) {
      if (t + 3 < cT)
        issue_bundle_tdm(gB + (size_t)(t + 3) * RS, &stage[(t + 3) & 7][0], lane);
      if (t < 3) {
        wait_async_bundles(2 - t);               // bundle t came via async path
      } else {
        int rem = cT - 1 - t;
        wait_tensor_bundles(rem > 3 ? 3 : rem);  // bundle t came via TDM
      }
    }
    __syncthreads();

    const float* cur = &stage[t & 7][0];

    // ---- c[u]: positive-duration arrivals (+ init at (0,0)) ----
    float c = (t == 0 && u == 0) ? 0.f : NEG_INF;
    if (u < cU) {
#pragma unroll
      for (int i = 1; i < cD; ++i) {             // durations[i] == i
        const int ts = t - i;
        if (ts >= 0) {
          const float* sp = &stage[ts & 7][0];
          const float* ap = &alpha_s[ts % 5][0];
          float stay = ap[u] + sp[SB_OFF + 4 * u + (i - 1)];
          float move = NEG_INF;
          if (u > 0) move = ap[u - 1] + sp[SY_OFF + 4 * (u - 1) + (i - 1)];
          c = lax(c, lax(stay, move));
        }
      }
    }

    // ---- scan 1: exclusive cumsum of e0 along u -> S ----
    float e0u = (u < cU) ? cur[E0_OFF + u] : 0.f;
    sc[0][tid] = e0u;
    __syncthreads();
    int s = 0;
#pragma unroll
    for (int off = 1; off < 128; off <<= 1) {
      float v = sc[s][tid];
      if (tid >= off) v += sc[s][tid - off];
      sc[s ^ 1][tid] = v;
      s ^= 1;
      __syncthreads();
    }
    const float S = (tid > 0) ? sc[s][tid - 1] : 0.f;

    // ---- scan 2: inclusive cum-logaddexp of (c - S) ----
    float g = (u < cU) ? (c - S) : NEG_INF;
    sc[0][tid] = g;
    __syncthreads();
    s = 0;
#pragma unroll
    for (int off = 1; off < 128; off <<= 1) {
      float v = sc[s][tid];
      if (tid >= off) v = lax(v, sc[s][tid - off]);
      sc[s ^ 1][tid] = v;
      s ^= 1;
      __syncthreads();
    }
    const float row = S + sc[s][tid];

    if (u < cU) {
      alpha_s[t % 5][u] = row;
      // final-term capture: alphas[tt,L] + blank_lp[tt,L] + dlp[tt,L,i] == row + sb
      if (u == Lb) {
#pragma unroll
        for (int i = 1; i < cD; ++i) {
          const int tt  = alen - i;
          int ttc = tt < 0 ? 0 : (tt > cT - 1 ? cT - 1 : tt);
          if (tt >= 0 && t == ttc)
            term[i - 1] = row + cur[SB_OFF + 4 * u + (i - 1)];
        }
      }
    }
    __syncthreads();   // protects slot (t+4)&7 reuse next iteration
  }

  if (tid == 0) {
    const float ll = lax(lax(term[0], term[1]), lax(term[2], term[3]));
    partial[b] = -ll;
  }
}

// =====================================================================
// Kernel C: mean over batch of -loglike -> single output element.
// =====================================================================
__global__ void tdt_final(const float* __restrict__ partial,
                          float* __restrict__ out) {
  out[0] = (partial[0] + partial[1] + partial[2] + partial[3]) * (1.0f / (float)cB);
}

// =====================================================================
// Launch
// =====================================================================
extern "C" void kernel_launch(void* const* d_in, const int* in_sizes, int n_in,
                              void* d_out, int out_size, void* d_ws, size_t ws_size,
                              hipStream_t stream) {
  (void)in_sizes; (void)n_in; (void)out_size; (void)ws_size;
  const float* acts       = (const float*)d_in[0];
  const int*   labels     = (const int*)d_in[1];
  const int*   act_lens   = (const int*)d_in[2];
  const int*   label_lens = (const int*)d_in[3];

  float* wsb     = (float*)d_ws;                       // bundles: B*T*RS floats (4 MB)
  float* partial = wsb + (size_t)cB * cT * RS;         // per-batch -loglike (4 floats)

  tdt_softmax_pack<<<dim3(1024), dim3(256), 0, stream>>>(acts, labels, wsb);
  tdt_dp<<<dim3(cB), dim3(128), 0, stream>>>(wsb, act_lens, label_lens, partial);
  tdt_final<<<dim3(1), dim3(1), 0, stream>>>(partial, (float*)d_out);
}